// EdgeSA_25735444037757
// MI455X (gfx1250) — compile-verified
//
#include <hip/hip_runtime.h>
#include <hip/hip_bf16.h>
#include <math.h>

// Problem constants (from setup_inputs)
#define B_    8
#define C_    128
#define N_    16384
#define M_    1024
#define KNN   16
#define NG_   4
#define CPG   32          // C_ / NG_
#define EPS_  1e-5f
#define SLOPE_ 0.2f

typedef float v2f __attribute__((ext_vector_type(2)));
typedef float v8f __attribute__((ext_vector_type(8)));

// ---------------------------------------------------------------------------
// 1) Farthest point sampling: one block per batch, 256 threads.
//    dists[] lives in global workspace (L2 resident, 64KB per batch).
// ---------------------------------------------------------------------------
__global__ __launch_bounds__(256) void fps_kernel(const float* __restrict__ pos,
                                                  float* __restrict__ dists,
                                                  int* __restrict__ fps_idx,
                                                  float* __restrict__ coor_out,
                                                  float* __restrict__ fpsf_out) {
    const int b   = blockIdx.x;
    const int tid = threadIdx.x;
    float* dist = dists + (size_t)b * N_;

    __shared__ float rv[256];
    __shared__ int   ri[256];
    __shared__ int   curs;

    for (int n = tid; n < N_; n += 256) dist[n] = 1e10f;
    __syncthreads();

    int cur = 0;
    for (int it = 0; it < M_; ++it) {
        const float cx = pos[((size_t)b * N_ + cur) * 3 + 0];
        const float cy = pos[((size_t)b * N_ + cur) * 3 + 1];
        const float cz = pos[((size_t)b * N_ + cur) * 3 + 2];
        if (tid == 0) {
            fps_idx[b * M_ + it]  = cur;
            fpsf_out[b * M_ + it] = (float)cur;
            coor_out[(b * M_ + it) * 3 + 0] = cx;
            coor_out[(b * M_ + it) * 3 + 1] = cy;
            coor_out[(b * M_ + it) * 3 + 2] = cz;
        }
        float lv = -1.0f;
        int   li = 0;
        for (int n = tid; n < N_; n += 256) {
            const float dx = pos[((size_t)b * N_ + n) * 3 + 0] - cx;
            const float dy = pos[((size_t)b * N_ + n) * 3 + 1] - cy;
            const float dz = pos[((size_t)b * N_ + n) * 3 + 2] - cz;
            const float d  = dx * dx + dy * dy + dz * dz;
            const float nd = fminf(dist[n], d);
            dist[n] = nd;
            if (nd > lv) { lv = nd; li = n; }
        }
        rv[tid] = lv; ri[tid] = li;
        __syncthreads();
        for (int s = 128; s > 0; s >>= 1) {
            if (tid < s) {
                if (rv[tid + s] > rv[tid] ||
                    (rv[tid + s] == rv[tid] && ri[tid + s] < ri[tid])) {
                    rv[tid] = rv[tid + s];
                    ri[tid] = ri[tid + s];
                }
            }
            __syncthreads();
        }
        if (tid == 0) curs = ri[0];
        __syncthreads();
        cur = curs;
    }
}

// ---------------------------------------------------------------------------
// 2) kNN: one thread per (b, m) query, streaming scan with register top-16.
// ---------------------------------------------------------------------------
__global__ __launch_bounds__(256) void knn_kernel(const float* __restrict__ pos,
                                                  const int* __restrict__ fps_idx,
                                                  int* __restrict__ knn_idx) {
    const int t = blockIdx.x * 256 + threadIdx.x;
    if (t >= B_ * M_) return;
    const int b = t / M_;
    const int m = t % M_;
    const int q = fps_idx[b * M_ + m];
    const float qx = pos[((size_t)b * N_ + q) * 3 + 0];
    const float qy = pos[((size_t)b * N_ + q) * 3 + 1];
    const float qz = pos[((size_t)b * N_ + q) * 3 + 2];

    float bd[KNN];
    int   bi[KNN];
#pragma unroll
    for (int i = 0; i < KNN; ++i) { bd[i] = 3.0e38f; bi[i] = 0; }
    float wv = 3.0e38f;
    int   wslot = 0;

    const float* prow = pos + (size_t)b * N_ * 3;
    for (int n = 0; n < N_; ++n) {
        __builtin_prefetch(prow + (size_t)(n + 64) * 3, 0, 0);
        const float dx = prow[(size_t)n * 3 + 0] - qx;
        const float dy = prow[(size_t)n * 3 + 1] - qy;
        const float dz = prow[(size_t)n * 3 + 2] - qz;
        const float d  = dx * dx + dy * dy + dz * dz;
        if (d < wv) {
            bd[wslot] = d; bi[wslot] = n;
            wv = -1.0f;
#pragma unroll
            for (int i = 0; i < KNN; ++i)
                if (bd[i] > wv) { wv = bd[i]; wslot = i; }
        }
    }
#pragma unroll
    for (int i = 0; i < KNN; ++i) knn_idx[(t) * KNN + i] = bi[i];
}

// ---------------------------------------------------------------------------
// 3) Edge-feature gather + 1x1 conv GEMM via V_WMMA_F32_16X16X4_F32.
//    One block per (b, m): stage edge (256 x 16) in LDS; 8 waves each compute
//    a 16x16 output tile (16 rows of w1 x the 16 neighbors) with 64 WMMAs.
//    Also accumulates GroupNorm sum / sumsq per (b, group) via atomics.
// ---------------------------------------------------------------------------
__global__ __launch_bounds__(256) void edge_gemm_kernel(
        const float* __restrict__ feature, const float* __restrict__ w1,
        const int* __restrict__ fps_idx, const int* __restrict__ knn_idx,
        float* __restrict__ h, float* __restrict__ stats) {
    __shared__ float edge_s[2 * C_][KNN];   // 16 KB
    __shared__ int   nbr_s[KNN];
    __shared__ int   q_s;
    __shared__ float red1[256], red2[256];

    const int blk = blockIdx.x;
    const int b   = blk / M_;
    const int m   = blk % M_;
    const int tid = threadIdx.x;

    if (tid == 0) q_s = fps_idx[b * M_ + m];
    if (tid < KNN) nbr_s[tid] = knn_idx[(b * M_ + m) * KNN + tid];
    __syncthreads();
    const int q = q_s;

    {   // edge[c][k] = feat[nbr_k][c] - fq[c]  (c < 128)  |  fq[c-128]  (c >= 128)
        const int c = tid;
        if (c < C_) {
            const float* frow = feature + ((size_t)b * C_ + c) * N_;
            const float fq = frow[q];
#pragma unroll
            for (int k = 0; k < KNN; ++k) edge_s[c][k] = frow[nbr_s[k]] - fq;
        } else {
            const int c2 = c - C_;
            const float* frow = feature + ((size_t)b * C_ + c2) * N_;
            const float fq = frow[q];
#pragma unroll
            for (int k = 0; k < KNN; ++k) edge_s[c][k] = fq;
        }
    }
    __syncthreads();

    const int wave = tid >> 5;
    const int lane = tid & 31;
    const int half = lane >> 4;     // 0: K=0,1  1: K=2,3 (ISA 16x4 f32 A layout)
    const int l16  = lane & 15;

    const float* arow = w1 + (size_t)(wave * 16 + l16) * (2 * C_) + 2 * half;

    v8f acc = {};
    for (int cc = 0; cc < 2 * C_; cc += 4) {
        v2f a, bm;
        a.x  = arow[cc];
        a.y  = arow[cc + 1];
        bm.x = edge_s[cc + 2 * half][l16];
        bm.y = edge_s[cc + 2 * half + 1][l16];
        // D = A(16x4 f32) * B(4x16 f32) + C ; emits v_wmma_f32_16x16x4_f32
        acc = __builtin_amdgcn_wmma_f32_16x16x4_f32(
                /*neg_a=*/false, a, /*neg_b=*/false, bm,
                /*c_mod=*/(short)0, acc, /*reuse_a=*/false, /*reuse_b=*/false);
    }

    // C/D layout: VGPR r -> rows r (lanes 0-15) / r+8 (lanes 16-31), col = lane%16
    float s1 = 0.f, s2 = 0.f;
#pragma unroll
    for (int r = 0; r < 8; ++r) {
        const int o = wave * 16 + 8 * half + r;
        const float v = acc[r];
        h[(((size_t)b * C_ + o) * M_ + m) * KNN + l16] = v;
        s1 += v;
        s2 += v * v;
    }

    red1[tid] = s1; red2[tid] = s2;
    __syncthreads();
    if (lane == 0) {
        float t1 = 0.f, t2 = 0.f;
        for (int i = 0; i < 32; ++i) { t1 += red1[wave * 32 + i]; t2 += red2[wave * 32 + i]; }
        const int g = wave >> 1;   // wave's 16 rows sit inside one 32-channel group
        atomicAdd(&stats[(b * NG_ + g) * 2 + 0], t1);
        atomicAdd(&stats[(b * NG_ + g) * 2 + 1], t2);
    }
}

// ---------------------------------------------------------------------------
// 4) GroupNorm apply + LeakyReLU + max over K -> out (B, C, M)
// ---------------------------------------------------------------------------
__global__ __launch_bounds__(256) void norm_max_kernel(
        const float* __restrict__ h, const float* __restrict__ stats,
        const float* __restrict__ gamma, const float* __restrict__ beta,
        float* __restrict__ out) {
    const int t = blockIdx.x * 256 + threadIdx.x;
    if (t >= B_ * C_ * M_) return;
    const int b   = t / (C_ * M_);
    const int rem = t % (C_ * M_);
    const int c   = rem / M_;
    const int m   = rem % M_;
    const int g   = c / CPG;

    const float s1  = stats[(b * NG_ + g) * 2 + 0];
    const float s2  = stats[(b * NG_ + g) * 2 + 1];
    const float cnt = (float)(CPG * M_ * KNN);
    const float mean = s1 / cnt;
    const float var  = s2 / cnt - mean * mean;
    const float rs   = rsqrtf(var + EPS_);
    const float ga = gamma[c], be = beta[c];

    const float* hp = h + (((size_t)b * C_ + c) * M_ + m) * KNN;
    float mx = -3.4e38f;
#pragma unroll
    for (int k = 0; k < KNN; ++k) {
        float v = (hp[k] - mean) * rs * ga + be;
        v = v > 0.f ? v : SLOPE_ * v;
        mx = fmaxf(mx, v);
    }
    out[t] = mx;   // t == b*C*M + c*M + m
}

// ---------------------------------------------------------------------------
extern "C" void kernel_launch(void* const* d_in, const int* in_sizes, int n_in,
                              void* d_out, int out_size, void* d_ws, size_t ws_size,
                              hipStream_t stream) {
    (void)in_sizes; (void)n_in; (void)out_size; (void)ws_size;

    const float* feature = (const float*)d_in[0];   // (B, C, N)
    const float* pos     = (const float*)d_in[1];   // (B, N, 3)
    const float* w1      = (const float*)d_in[2];   // (C, 2C)
    const float* gamma   = (const float*)d_in[3];   // (C,)
    const float* beta    = (const float*)d_in[4];   // (C,)
    // d_in[5] = num (device scalar); fixed to M_ = 1024 per setup_inputs

    float* out      = (float*)d_out;                        // (B, C, M)
    float* coor_out = out + (size_t)B_ * C_ * M_;           // (B, M, 3)
    float* fpsf_out = coor_out + (size_t)B_ * M_ * 3;       // (B, M) as float

    char* ws = (char*)d_ws;
    size_t off = 0;
    float* h        = (float*)(ws + off); off += (size_t)B_ * C_ * M_ * KNN * sizeof(float); // 64 MB
    float* dists    = (float*)(ws + off); off += (size_t)B_ * N_ * sizeof(float);            // 512 KB
    int*   fps_idx  = (int*)  (ws + off); off += (size_t)B_ * M_ * sizeof(int);              // 32 KB
    int*   knn_idx  = (int*)  (ws + off); off += (size_t)B_ * M_ * KNN * sizeof(int);        // 512 KB
    float* stats    = (float*)(ws + off); // B*NG*2 floats

    hipMemsetAsync(stats, 0, (size_t)B_ * NG_ * 2 * sizeof(float), stream);

    fps_kernel<<<B_, 256, 0, stream>>>(pos, dists, fps_idx, coor_out, fpsf_out);
    knn_kernel<<<(B_ * M_ + 255) / 256, 256, 0, stream>>>(pos, fps_idx, knn_idx);
    edge_gemm_kernel<<<B_ * M_, 256, 0, stream>>>(feature, w1, fps_idx, knn_idx, h, stats);
    norm_max_kernel<<<(B_ * C_ * M_ + 255) / 256, 256, 0, stream>>>(h, stats, gamma, beta, out);
}